// LiftSplatShootEncoder_37941741093390
// MI455X (gfx1250) — compile-verified
//
#include <hip/hip_runtime.h>

typedef __attribute__((ext_vector_type(2))) float v2f;
typedef __attribute__((ext_vector_type(8))) float v8f;

#define B_      4
#define N_      4
#define D_      41
#define FH      16
#define FW      44
#define C_      64
#define SLICE   (D_*FH*FW)      // 28864 points per (b,n), divisible by 16
#define NPTS    (B_*N_*SLICE)   // 461824
#define NX      256
#define NY      256
#define OUT_ELEMS (B_*C_*NY*NX) // 16777216

// ---------------------------------------------------------------------------
// 3x3 inverse via adjugate
// ---------------------------------------------------------------------------
__device__ inline void inv3(const float* m, float* o) {
  float a=m[0],b=m[1],c=m[2],d=m[3],e=m[4],f=m[5],g=m[6],h=m[7],i=m[8];
  float A =  (e*i - f*h);
  float Bm = -(d*i - f*g);
  float Cm =  (d*h - e*g);
  float det = a*A + b*Bm + c*Cm;
  float r = 1.0f / det;
  o[0] =  A*r;           o[1] = -(b*i - c*h)*r; o[2] =  (b*f - c*e)*r;
  o[3] =  Bm*r;          o[4] =  (a*i - c*g)*r; o[5] = -(a*f - c*d)*r;
  o[6] =  Cm*r;          o[7] = -(a*h - b*g)*r; o[8] =  (a*e - b*d)*r;
}

// ---------------------------------------------------------------------------
// Per-(b,n): M1 = [inv(post_rots) | -inv(post_rots)*post_trans]  (3x4)
//            M2 = [rots*inv(intrins) | trans]                    (3x4)
// 24 floats per (b,n), 16 pairs.
// ---------------------------------------------------------------------------
__global__ void prep_kernel(const float* __restrict__ rots,
                            const float* __restrict__ trans,
                            const float* __restrict__ intrins,
                            const float* __restrict__ post_rots,
                            const float* __restrict__ post_trans,
                            float* __restrict__ mats) {
  const int t = threadIdx.x;
  if (t >= B_*N_) return;
  const float* R  = rots      + t*9;
  const float* K  = intrins   + t*9;
  const float* PR = post_rots + t*9;
  const float* T  = trans      + t*3;
  const float* PT = post_trans + t*3;
  float iPR[9], iK[9], CB[9];
  inv3(PR, iPR);
  inv3(K,  iK);
  for (int i = 0; i < 3; ++i)
    for (int j = 0; j < 3; ++j) {
      float acc = 0.f;
      for (int k = 0; k < 3; ++k) acc += R[i*3+k] * iK[k*3+j];
      CB[i*3+j] = acc;
    }
  float* M1 = mats + t*24;
  float* M2 = M1 + 12;
  for (int i = 0; i < 3; ++i) {
    M1[i*4+0] = iPR[i*3+0]; M1[i*4+1] = iPR[i*3+1]; M1[i*4+2] = iPR[i*3+2];
    M1[i*4+3] = -(iPR[i*3+0]*PT[0] + iPR[i*3+1]*PT[1] + iPR[i*3+2]*PT[2]);
    M2[i*4+0] = CB[i*3+0];  M2[i*4+1] = CB[i*3+1];  M2[i*4+2] = CB[i*3+2];
    M2[i*4+3] = T[i];
  }
}

// ---------------------------------------------------------------------------
// Zero the 64MB output (float4 stores)
// ---------------------------------------------------------------------------
__global__ __launch_bounds__(256) void zero_kernel(float4* __restrict__ out) {
  out[(size_t)blockIdx.x * blockDim.x + threadIdx.x] = make_float4(0.f, 0.f, 0.f, 0.f);
}

// ---------------------------------------------------------------------------
// Geometry: each wave32 transforms 16 points with two V_WMMA_F32_16X16X4_F32.
//   Stage 1: D1 = A1(16x4) x B1(4x16);  A1 rows 0-2 = M1 (dup at rows 8-10),
//            B1 column n = [fx, fy, fz, 1] of point n.
//   Elementwise: (X*Z, Y*Z, Z, 1) — lane-local thanks to the row duplication.
//   Stage 2: D2 = A2 x B2 with A2 = M2.
// All waves fully active (grid sized exactly), satisfying WMMA EXEC==all-1s.
// ---------------------------------------------------------------------------
__global__ __launch_bounds__(256) void geom_kernel(const float* __restrict__ mats,
                                                   int* __restrict__ flat) {
  const int gtid = blockIdx.x * blockDim.x + threadIdx.x;
  const int wave = gtid >> 5;
  const int lane = threadIdx.x & 31;
  const int half = lane >> 4;       // which 16-lane half
  const int r    = lane & 15;       // row (A) / column (B,D) index
  const int base = wave << 4;       // first point of this 16-point group
  const int s    = base / SLICE;    // (b,n) pair id, wave-uniform

  const float* __restrict__ M1 = mats + s * 24;
  const float* __restrict__ M2 = M1 + 12;

  // A-matrix fill: lane holds A row (lane&15); v0 = K(2*half), v1 = K(2*half+1).
  // Rows 0-2 carry the 3x4 transform; rows 8-10 duplicate it so lanes 16-31
  // of D also see the point coordinates. Other rows are zero.
  const bool isrow = (r < 3) || (r >= 8 && r < 11);
  const int  mrow  = isrow ? ((r < 3) ? r : (r - 8)) : 0;
  const float fm   = isrow ? 1.0f : 0.0f;
  const int   k0   = half * 2;

  v2f a1; a1.x = M1[mrow*4 + k0] * fm; a1.y = M1[mrow*4 + k0 + 1] * fm;
  v2f a2; a2.x = M2[mrow*4 + k0] * fm; a2.y = M2[mrow*4 + k0 + 1] * fm;

  // Frustum coords of point (base + r): p -> (d,h,w)
  const int p  = base + r;
  const int q  = p % SLICE;
  const int dd = q / (FH*FW);
  const int rm = q % (FH*FW);
  const int hh = rm / FW;
  const int ww = rm % FW;
  const float fx = (float)ww * (703.0f / 43.0f);  // linspace(0,703,44)
  const float fy = (float)hh * 17.0f;             // linspace(0,255,16)
  const float fz = 4.0f + (float)dd;              // arange(4,45,1)

  // B1 (4x16): v0 = row K0 (lanes 0-15) / K2 (lanes 16-31); v1 = K1 / K3.
  v2f b1; b1.x = half ? fz : fx; b1.y = half ? 1.0f : fy;
  v8f c0 = {};

  v8f d1 = __builtin_amdgcn_wmma_f32_16x16x4_f32(
      false, a1, false, b1, (short)0, c0, false, false);
  const float X = d1[0], Y = d1[1], Z = d1[2];  // coords of point (base+r), both halves

  // pts = [X*Z, Y*Z, Z, 1] -> B2
  v2f b2; b2.x = half ? Z : X * Z; b2.y = half ? 1.0f : Y * Z;

  v8f d2 = __builtin_amdgcn_wmma_f32_16x16x4_f32(
      false, a2, false, b2, (short)0, c0, false, false);
  const float gx = d2[0], gy = d2[1], gz = d2[2];

  // Voxelize: int cast truncates toward zero, matching .astype(int32)
  const int cx = (int)((gx + 51.2f) / 0.4f);
  const int cy = (int)((gy + 51.2f) / 0.4f);
  const int cz = (int)(gz / 48.0f);
  const bool kept = (cx >= 0) && (cx < NX) && (cy >= 0) && (cy < NY) && (cz == 0);
  const int bidx = s >> 2;  // b = s / N_
  const int fl = kept ? ((bidx << 16) | (cy * NX + cx)) : -1;

  if (lane < 16) flat[p] = fl;
}

// ---------------------------------------------------------------------------
// Scatter-add: one thread per (point, channel). x reads fully coalesced and
// non-temporal (streamed once); atomics land in L2 where the 64MB output lives.
// ---------------------------------------------------------------------------
__global__ __launch_bounds__(256) void scatter_kernel(const float* __restrict__ x,
                                                      const int* __restrict__ flat,
                                                      float* __restrict__ out) {
  const long long tid = (long long)blockIdx.x * blockDim.x + threadIdx.x;
  const int p  = (int)(tid >> 6);
  const int ch = (int)(tid & 63);
  const int f  = flat[p];          // wave-uniform (32 lanes share a point)
  if (f < 0) return;
  const float v = __builtin_nontemporal_load(x + tid);
  const int b  = f >> 16;
  const int sp = f & 0xFFFF;       // cy*NX + cx
  atomicAdd(out + (((size_t)(b * C_ + ch)) << 16) + (size_t)sp, v);
}

// ---------------------------------------------------------------------------
extern "C" void kernel_launch(void* const* d_in, const int* in_sizes, int n_in,
                              void* d_out, int out_size, void* d_ws, size_t ws_size,
                              hipStream_t stream) {
  const float* x          = (const float*)d_in[0];
  const float* rots       = (const float*)d_in[1];
  const float* trans      = (const float*)d_in[2];
  const float* intrins    = (const float*)d_in[3];
  const float* post_rots  = (const float*)d_in[4];
  const float* post_trans = (const float*)d_in[5];
  float* out  = (float*)d_out;
  float* mats = (float*)d_ws;                    // 16*24 floats
  int*   flat = (int*)((char*)d_ws + 2048);      // NPTS ints (~1.8 MB)

  prep_kernel<<<1, 32, 0, stream>>>(rots, trans, intrins, post_rots, post_trans, mats);
  zero_kernel<<<OUT_ELEMS / 4 / 256, 256, 0, stream>>>((float4*)out);
  geom_kernel<<<(NPTS / 16) / 8, 256, 0, stream>>>(mats, flat);   // 8 waves/block
  scatter_kernel<<<(NPTS * C_) / 256, 256, 0, stream>>>(x, flat, out);
}